// GNN_78237124263951
// MI455X (gfx1250) — compile-verified
//
#include <hip/hip_runtime.h>

typedef __attribute__((ext_vector_type(2))) float v2f;
typedef __attribute__((ext_vector_type(8))) float v8f;
typedef __attribute__((ext_vector_type(4))) int   v4i;

#define NNODES  50000
#define NEDGES  600000
#define DFEAT   128     // D
#define CMID    256     // 2D
#define NBOND   8       // BV
#define NLAYERS 5
#define MTILES  3125    // NNODES/16

// ---------------------------------------------------------------------------
// Async global->LDS staging (gfx1250 ASYNCcnt path) with safe fallback.
// Probe result: builtin exists; param0 is int4* in addrspace(1), param1 in
// addrspace(3); two integer immediates (offset, cpol).
// ---------------------------------------------------------------------------
#if __has_builtin(__builtin_amdgcn_global_load_async_to_lds_b128) && \
    __has_builtin(__builtin_amdgcn_s_wait_asynccnt)
#define HAVE_ASYNC_LDS 1
#endif

typedef __attribute__((address_space(1))) v4i gv4i;
typedef __attribute__((address_space(3))) v4i lv4i;

__device__ __forceinline__ void copy16_to_lds(const float* __restrict__ g,
                                              float* lds) {
#ifdef HAVE_ASYNC_LDS
  __builtin_amdgcn_global_load_async_to_lds_b128((gv4i*)g, (lv4i*)lds, 0, 0);
#else
  *reinterpret_cast<float4*>(lds) = *reinterpret_cast<const float4*>(g);
#endif
}

__device__ __forceinline__ void lds_stage_fence() {
#ifdef HAVE_ASYNC_LDS
  __builtin_amdgcn_s_wait_asynccnt(0);
#endif
  __syncthreads();
}

// ---------------------------------------------------------------------------
// AtomEncoder: h[n,:] = sum_k atom_emb[k][x[n,k]][:]
// ---------------------------------------------------------------------------
__global__ void atom_encoder(const int* __restrict__ x,
                             const float* __restrict__ atom_emb,
                             float* __restrict__ h) {
  int idx = blockIdx.x * blockDim.x + threadIdx.x;
  int n = idx >> 5;
  if (n >= NNODES) return;
  int d0 = (idx & 31) * 4;
  float4 acc = make_float4(0.f, 0.f, 0.f, 0.f);
#pragma unroll
  for (int k = 0; k < 9; ++k) {
    int a = x[n * 9 + k];
    const float4 v =
        *reinterpret_cast<const float4*>(&atom_emb[(k * 128 + a) * DFEAT + d0]);
    acc.x += v.x; acc.y += v.y; acc.z += v.z; acc.w += v.w;
  }
  *reinterpret_cast<float4*>(&h[n * DFEAT + d0]) = acc;
}

// ---------------------------------------------------------------------------
// Message + scatter-add: agg[dst] += relu(h[src] + ee)
// bond table (12 KB) async-staged to LDS; one wave per edge, lane = 4 floats
// ---------------------------------------------------------------------------
__global__ void scatter_msgs(const float* __restrict__ h,
                             const int* __restrict__ edge_index,
                             const int* __restrict__ edge_attr,
                             const float* __restrict__ bond_emb_l,
                             float* __restrict__ agg) {
  __shared__ __align__(16) float lds_bond[3 * NBOND * DFEAT];  // 12 KB
  for (int i = threadIdx.x; i < (3 * NBOND * DFEAT) / 4; i += blockDim.x)
    copy16_to_lds(&bond_emb_l[i * 4], &lds_bond[i * 4]);
  lds_stage_fence();

  int lane  = threadIdx.x & 31;
  int wave  = (blockIdx.x * blockDim.x + threadIdx.x) >> 5;
  int nwave = (gridDim.x * blockDim.x) >> 5;
  int d0 = lane * 4;

  for (int e = wave; e < NEDGES; e += nwave) {
    int src = edge_index[e];
    int dst = edge_index[NEDGES + e];
    int a0 = edge_attr[e * 3 + 0];
    int a1 = edge_attr[e * 3 + 1];
    int a2 = edge_attr[e * 3 + 2];
    float4 hv = *reinterpret_cast<const float4*>(&h[src * DFEAT + d0]);
    float4 e0 = *reinterpret_cast<const float4*>(&lds_bond[(0 * NBOND + a0) * DFEAT + d0]);
    float4 e1 = *reinterpret_cast<const float4*>(&lds_bond[(1 * NBOND + a1) * DFEAT + d0]);
    float4 e2 = *reinterpret_cast<const float4*>(&lds_bond[(2 * NBOND + a2) * DFEAT + d0]);
    float m0 = fmaxf(hv.x + e0.x + e1.x + e2.x, 0.f);
    float m1 = fmaxf(hv.y + e0.y + e1.y + e2.y, 0.f);
    float m2 = fmaxf(hv.z + e0.z + e1.z + e2.z, 0.f);
    float m3 = fmaxf(hv.w + e0.w + e1.w + e2.w, 0.f);
    float* ap = &agg[dst * DFEAT + d0];
    unsafeAtomicAdd(ap + 0, m0);
    unsafeAtomicAdd(ap + 1, m1);
    unsafeAtomicAdd(ap + 2, m2);
    unsafeAtomicAdd(ap + 3, m3);
  }
}

// ---------------------------------------------------------------------------
// GEMM1: hmid[N,256] = ((1+eps)*h + agg) @ W1 + b1      (WMMA f32 16x16x4)
// Block: 8 waves = 128 rows x 64 cols. B tile (128K x 64) staged in LDS,
// row stride padded to 72 floats so half-waves hit disjoint bank groups.
// Each wave: 16 rows x 4 col-tiles (acc[4]).
// ---------------------------------------------------------------------------
#define G_COLS   64
#define G_STRIDE 72

__global__ void gemm1_wmma(const float* __restrict__ h,
                           const float* __restrict__ agg,
                           const float* __restrict__ W1l,
                           const float* __restrict__ b1l,
                           const float* __restrict__ eps_l,
                           float* __restrict__ hmid) {
  __shared__ __align__(16) float ldsB[DFEAT * G_STRIDE];  // 36 KB
  int tid = threadIdx.x;
  int c0 = blockIdx.y * G_COLS;

  for (int i = tid; i < DFEAT * (G_COLS / 4); i += 256) {
    int k = i >> 4;            // / (G_COLS/4)
    int c = (i & 15) * 4;
    copy16_to_lds(&W1l[k * CMID + c0 + c], &ldsB[k * G_STRIDE + c]);
  }
  lds_stage_fence();

  int wave = tid >> 5, lane = tid & 31;
  int mt = blockIdx.x * 8 + wave;
  int lr = lane & 15, hi = lane >> 4;
  if (mt >= MTILES) return;

  int row0 = mt * 16;
  float heps = 1.0f + eps_l[0];
  const float* Ah = h   + (row0 + lr) * DFEAT;
  const float* Ag = agg + (row0 + lr) * DFEAT;

  v8f acc[4] = {};
#pragma unroll 2
  for (int k = 0; k < DFEAT; k += 4) {
    int ka = k + hi * 2;
    v2f a;
    a.x = fmaf(heps, Ah[ka + 0], Ag[ka + 0]);
    a.y = fmaf(heps, Ah[ka + 1], Ag[ka + 1]);
    const float* B0 = &ldsB[ka * G_STRIDE + lr];
    const float* B1 = B0 + G_STRIDE;
#pragma unroll
    for (int nt = 0; nt < 4; ++nt) {
      v2f b;
      b.x = B0[nt * 16];
      b.y = B1[nt * 16];
      acc[nt] = __builtin_amdgcn_wmma_f32_16x16x4_f32(false, a, false, b,
                                                      (short)0, acc[nt],
                                                      false, false);
    }
  }

  int rbase = row0 + hi * 8;
#pragma unroll
  for (int nt = 0; nt < 4; ++nt) {
    int col = c0 + nt * 16 + lr;
    float bias = b1l[col];
#pragma unroll
    for (int r = 0; r < 8; ++r)
      hmid[(rbase + r) * CMID + col] = acc[nt][r] + bias;
  }
}

// ---------------------------------------------------------------------------
// GEMM2: z2[N,128] = relu(hmid*scale + shift) @ W2 + b2  (WMMA f32 16x16x4)
// K=256 double-staged in two 128-row LDS phases; scale/shift cached in LDS.
// ---------------------------------------------------------------------------
__global__ void gemm2_wmma(const float* __restrict__ hmid,
                           const float* __restrict__ scale,
                           const float* __restrict__ shift,
                           const float* __restrict__ W2l,
                           const float* __restrict__ b2l,
                           float* __restrict__ z2) {
  __shared__ __align__(16) float ldsB[128 * G_STRIDE];  // 36 KB (one K-phase)
  __shared__ float ldsScale[CMID], ldsShift[CMID];
  int tid = threadIdx.x;
  int c0 = blockIdx.y * G_COLS;

  int wave = tid >> 5, lane = tid & 31;
  int mt = blockIdx.x * 8 + wave;
  int lr = lane & 15, hi = lane >> 4;
  bool valid = (mt < MTILES);
  int row0 = mt * 16;
  const float* Arow = hmid + (row0 + lr) * CMID;

  v8f acc[4] = {};
  for (int kc = 0; kc < 2; ++kc) {
    int kbase = kc * 128;
    for (int i = tid; i < 128 * (G_COLS / 4); i += 256) {
      int k = i >> 4;
      int c = (i & 15) * 4;
      copy16_to_lds(&W2l[(kbase + k) * DFEAT + c0 + c], &ldsB[k * G_STRIDE + c]);
    }
    if (kc == 0) {
      ldsScale[tid] = scale[tid];
      ldsShift[tid] = shift[tid];
    }
    lds_stage_fence();

    if (valid) {
#pragma unroll 2
      for (int k = 0; k < 128; k += 4) {
        int kl = k + hi * 2;       // LDS row
        int ka = kbase + kl;       // global K for A / scale / shift
        v2f a;
        a.x = fmaxf(fmaf(Arow[ka + 0], ldsScale[ka + 0], ldsShift[ka + 0]), 0.f);
        a.y = fmaxf(fmaf(Arow[ka + 1], ldsScale[ka + 1], ldsShift[ka + 1]), 0.f);
        const float* B0 = &ldsB[kl * G_STRIDE + lr];
        const float* B1 = B0 + G_STRIDE;
#pragma unroll
        for (int nt = 0; nt < 4; ++nt) {
          v2f b;
          b.x = B0[nt * 16];
          b.y = B1[nt * 16];
          acc[nt] = __builtin_amdgcn_wmma_f32_16x16x4_f32(false, a, false, b,
                                                          (short)0, acc[nt],
                                                          false, false);
        }
      }
    }
    __syncthreads();  // protect LDS before restaging / exit
  }

  if (!valid) return;
  int rbase = row0 + hi * 8;
#pragma unroll
  for (int nt = 0; nt < 4; ++nt) {
    int col = c0 + nt * 16 + lr;
    float bias = b2l[col];
#pragma unroll
    for (int r = 0; r < 8; ++r)
      z2[(rbase + r) * DFEAT + col] = acc[nt][r] + bias;
  }
}

// ---------------------------------------------------------------------------
// Column sums / sum-of-squares for BatchNorm stats (biased, over N rows)
// ---------------------------------------------------------------------------
__global__ void colstats(const float* __restrict__ X,
                         float* __restrict__ sum, float* __restrict__ sumsq,
                         int C, int rowsPerBlock) {
  int c  = threadIdx.x % C;
  int g  = threadIdx.x / C;
  int ng = blockDim.x / C;
  int r0 = blockIdx.x * rowsPerBlock;
  float s = 0.f, ss = 0.f;
  for (int r = r0 + g; r < r0 + rowsPerBlock; r += ng) {
    float v = X[r * C + c];
    s += v;
    ss += v * v;
  }
  unsafeAtomicAdd(&sum[c], s);
  unsafeAtomicAdd(&sumsq[c], ss);
}

// Fold BN affine: scale = gamma*rsqrt(var+1e-5), shift = beta - mu*scale
__global__ void finalize_stats(const float* __restrict__ sum,
                               const float* __restrict__ sumsq,
                               const float* __restrict__ gam,
                               const float* __restrict__ bet,
                               float* __restrict__ scale,
                               float* __restrict__ shift,
                               int C, float invN) {
  int c = blockIdx.x * blockDim.x + threadIdx.x;
  if (c >= C) return;
  float mu  = sum[c] * invN;
  float var = sumsq[c] * invN - mu * mu;
  float inv = rsqrtf(var + 1e-5f);
  float sc  = inv * gam[c];
  scale[c] = sc;
  shift[c] = bet[c] - mu * sc;
}

// Outer BN (+ optional relu): Y = act(X*scale + shift), C = 128
__global__ void bn_act(const float* __restrict__ X,
                       const float* __restrict__ scale,
                       const float* __restrict__ shift,
                       float* __restrict__ Y, int doRelu) {
  int idx = blockIdx.x * blockDim.x + threadIdx.x;
  int n = idx >> 5;
  if (n >= NNODES) return;
  int d0 = (idx & 31) * 4;
  float4 v  = *reinterpret_cast<const float4*>(&X[n * DFEAT + d0]);
  float4 sc = *reinterpret_cast<const float4*>(&scale[d0]);
  float4 sh = *reinterpret_cast<const float4*>(&shift[d0]);
  float4 y;
  y.x = fmaf(v.x, sc.x, sh.x);
  y.y = fmaf(v.y, sc.y, sh.y);
  y.z = fmaf(v.z, sc.z, sh.z);
  y.w = fmaf(v.w, sc.w, sh.w);
  if (doRelu) {
    y.x = fmaxf(y.x, 0.f); y.y = fmaxf(y.y, 0.f);
    y.z = fmaxf(y.z, 0.f); y.w = fmaxf(y.w, 0.f);
  }
  *reinterpret_cast<float4*>(&Y[n * DFEAT + d0]) = y;
}

// ---------------------------------------------------------------------------
extern "C" void kernel_launch(void* const* d_in, const int* in_sizes, int n_in,
                              void* d_out, int out_size, void* d_ws, size_t ws_size,
                              hipStream_t stream) {
  const int*   x          = (const int*)d_in[0];
  const int*   edge_index = (const int*)d_in[1];
  const int*   edge_attr  = (const int*)d_in[2];
  const float* atom_emb   = (const float*)d_in[3];
  const float* bond_emb   = (const float*)d_in[4];
  const float* W1  = (const float*)d_in[5];
  const float* b1  = (const float*)d_in[6];
  const float* g1  = (const float*)d_in[7];
  const float* be1 = (const float*)d_in[8];
  const float* W2  = (const float*)d_in[9];
  const float* b2  = (const float*)d_in[10];
  const float* eps = (const float*)d_in[11];
  const float* g_out  = (const float*)d_in[12];
  const float* be_out = (const float*)d_in[13];

  char* ws = (char*)d_ws;
  float* h    = (float*)(ws + 0);            // N*128
  float* agg  = (float*)(ws + 25600000);     // N*128
  float* hmid = (float*)(ws + 51200000);     // N*256
  float* z2   = (float*)(ws + 102400000);    // N*128
  float* stat = (float*)(ws + 128000000);
  float* sum    = stat;
  float* sumsq  = stat + 256;
  float* scale1 = stat + 512;
  float* shift1 = stat + 768;
  float* scale2 = stat + 1024;
  float* shift2 = stat + 1280;

  const float invN = 1.0f / (float)NNODES;

  atom_encoder<<<(NNODES * 32 + 255) / 256, 256, 0, stream>>>(x, atom_emb, h);

  for (int l = 0; l < NLAYERS; ++l) {
    (void)hipMemsetAsync(agg, 0, (size_t)NNODES * DFEAT * sizeof(float), stream);
    scatter_msgs<<<2400, 256, 0, stream>>>(
        h, edge_index, edge_attr, bond_emb + (size_t)l * 3 * NBOND * DFEAT, agg);

    gemm1_wmma<<<dim3((MTILES + 7) / 8, CMID / G_COLS), 256, 0, stream>>>(
        h, agg, W1 + (size_t)l * DFEAT * CMID, b1 + (size_t)l * CMID, eps + l, hmid);

    (void)hipMemsetAsync(sum, 0, 512 * sizeof(float), stream);
    colstats<<<500, 256, 0, stream>>>(hmid, sum, sumsq, CMID, 100);
    finalize_stats<<<1, 256, 0, stream>>>(sum, sumsq, g1 + (size_t)l * CMID,
                                          be1 + (size_t)l * CMID,
                                          scale1, shift1, CMID, invN);

    gemm2_wmma<<<dim3((MTILES + 7) / 8, DFEAT / G_COLS), 256, 0, stream>>>(
        hmid, scale1, shift1, W2 + (size_t)l * CMID * DFEAT,
        b2 + (size_t)l * DFEAT, z2);

    (void)hipMemsetAsync(sum, 0, 512 * sizeof(float), stream);
    colstats<<<500, 256, 0, stream>>>(z2, sum, sumsq, DFEAT, 100);
    finalize_stats<<<1, 128, 0, stream>>>(sum, sumsq, g_out + (size_t)l * DFEAT,
                                          be_out + (size_t)l * DFEAT,
                                          scale2, shift2, DFEAT, invN);

    float* dst = (l == NLAYERS - 1) ? (float*)d_out : h;
    bn_act<<<(NNODES * 32 + 255) / 256, 256, 0, stream>>>(
        z2, scale2, shift2, dst, (l < NLAYERS - 1) ? 1 : 0);
  }
}